// ProtGramDirectGCN_10806137717551
// MI455X (gfx1250) — compile-verified
//
#include <hip/hip_runtime.h>
#include <hip/hip_bf16.h>

typedef __attribute__((ext_vector_type(2))) float v2f;
typedef __attribute__((ext_vector_type(8))) float v8f;
typedef __attribute__((ext_vector_type(4))) unsigned int u32x4;
typedef __attribute__((ext_vector_type(8))) int i32x8;
typedef __attribute__((ext_vector_type(4))) int i32x4;

#define DBLK 128            // feature dim
#define LRELU 0.01f

#if defined(__has_builtin)
#if __has_builtin(__builtin_amdgcn_tensor_load_to_lds) && __has_builtin(__builtin_amdgcn_s_wait_tensorcnt)
#define USE_TDM 1
#endif
#endif
#ifndef USE_TDM
#define USE_TDM 0
#endif

// ---------------------------------------------------------------------------
// h0 = x + pe (pe is [4,32] flattened -> 128 values broadcast over rows)
// ---------------------------------------------------------------------------
__global__ __launch_bounds__(256) void pe_add_kernel(const float* __restrict__ x,
                                                     const float* __restrict__ pe,
                                                     float* __restrict__ h, long long total) {
    long long g = (long long)blockIdx.x * blockDim.x + threadIdx.x;
    if (g >= total) return;
    h[g] = x[g] + pe[(int)(g & 127)];
}

// ---------------------------------------------------------------------------
// Wf = Wa + Wb   (fused main+shared weights, both layers at once)
// ---------------------------------------------------------------------------
__global__ __launch_bounds__(256) void wfuse_kernel(const float* __restrict__ wa,
                                                    const float* __restrict__ wb,
                                                    float* __restrict__ wf, int count) {
    int g = blockIdx.x * blockDim.x + threadIdx.x;
    if (g < count) wf[g] = wa[g] + wb[g];
}

// ---------------------------------------------------------------------------
// gcn[n,d] = constant[n,d] + Cin[n]*(b_mi[d]+b_si[d]) + Cout[n]*(b_mo[d]+b_so[d])
//            + Call[n]*b_u[d]
// (fully writes the accumulation buffer -> no memset needed)
// ---------------------------------------------------------------------------
__global__ __launch_bounds__(256) void gcn_init_kernel(const float* __restrict__ cnst,
        const float* __restrict__ cin, const float* __restrict__ cout_, const float* __restrict__ call,
        const float* __restrict__ b_mi, const float* __restrict__ b_si,
        const float* __restrict__ b_mo, const float* __restrict__ b_so,
        const float* __restrict__ b_u, float* __restrict__ gcn, long long total) {
    long long g = (long long)blockIdx.x * blockDim.x + threadIdx.x;
    if (g >= total) return;
    int node = (int)(g >> 7), d = (int)(g & 127);
    gcn[g] = cnst[g] + cin[node] * (b_mi[d] + b_si[d])
                     + cout_[node] * (b_mo[d] + b_so[d])
                     + call[node] * b_u[d];
}

// ---------------------------------------------------------------------------
// WMMA GEMM:  Y[nrows, NCOL] = H[nrows, K] @ W[NCOL, K]^T   (f32, 16x16x4 WMMA)
// MODE: 0 = plain, 1 = +bias, 2 = relu(+bias)
// 64 rows per 256-thread block (8 waves). H tile staged into LDS by the
// Tensor Data Mover (one tensor_load_to_lds per block, wave 0 issues it):
//   - pad_interval/pad_amount give the bank-conflict-free K+4 dword LDS stride
//   - tensor_dim1 = remaining rows -> OOB rows zero-filled by TDM (ragged tail)
// W is read from global (WGP$-resident, 64 KB hot, reused by every block).
// ---------------------------------------------------------------------------
template<int K, int NCOL, int MODE>
__global__ __launch_bounds__(256) void gemm_wmma_kernel(const float* __restrict__ H,
                                                        const float* __restrict__ W,
                                                        const float* __restrict__ bias,
                                                        float* __restrict__ Y, int nrows) {
    constexpr int LDA = K + 4;
    constexpr int NCT = NCOL / 16;       // column tiles
    constexpr int NTPW = (4 * NCT) / 8;  // tiles per wave (8 waves, 4 row-tiles)
    __shared__ float lds_h[64 * LDA];

    const int tid  = threadIdx.x;
    const int wave = tid >> 5;
    const int lane = tid & 31;
    const int row0 = blockIdx.x * 64;

#if USE_TDM
    // --- async DMA of the 64 x K f32 tile via the Tensor Data Mover ---
    if (tid < 32) {
        const unsigned lds_base = (unsigned)(uintptr_t)(void*)&lds_h[0];   // flat addr low 32b = LDS offset
        const unsigned long long ga =
            (unsigned long long)(uintptr_t)(const void*)(H + (size_t)row0 * K);
        const unsigned rows_rem = (unsigned)(nrows - row0);
        constexpr unsigned PAD_IVL = (K == 128) ? 6u : 5u;   // pad every 2^(c+1)=K dwords
        u32x4 g0;
        g0.x = 1u;                                           // count=1, user mode
        g0.y = lds_base;                                     // D#.lds_addr
        g0.z = (unsigned)(ga & 0xffffffffu);                 // D#.global_addr[31:0]
        g0.w = (unsigned)((ga >> 32) & 0x1ffffffu) | (2u << 30); // addr[56:32] | type=2
        i32x8 g1;
        g1[0] = (int)((2u << 16) | (1u << 20) | (PAD_IVL << 22) | (3u << 25));
                 // data_size=4B, pad_enable, pad_interval=K dwords, pad_amount=4 dwords
        g1[1] = (int)(((unsigned)K & 0xffffu) << 16);        // tensor_dim0[15:0] = K
        g1[2] = (int)((rows_rem & 0xffffu) << 16);           // dim0 hi=0 | tensor_dim1[15:0]
        g1[3] = (int)((rows_rem >> 16) | ((unsigned)K << 16)); // tensor_dim1 hi | tile_dim0=K
        g1[4] = 64;                                          // tile_dim1=64, tile_dim2=0
        g1[5] = K;                                           // tensor_dim0_stride[31:0] = K
        g1[6] = 0;
        g1[7] = 0;
        i32x4 zz = (i32x4){0, 0, 0, 0};
#if defined(__clang_major__) && (__clang_major__ >= 23)
        i32x8 z8 = (i32x8){0, 0, 0, 0, 0, 0, 0, 0};
        __builtin_amdgcn_tensor_load_to_lds(g0, g1, zz, zz, z8, 0);
#else
        __builtin_amdgcn_tensor_load_to_lds(g0, g1, zz, zz, 0);
#endif
        __builtin_amdgcn_s_wait_tensorcnt(0);
    }
    __syncthreads();
#else
    // --- fallback: manual staging (zero-pad ragged tail rows) ---
    constexpr int CHUNKS = 64 * (K / 4);
    for (int idx = tid; idx < CHUNKS; idx += 256) {
        int r  = idx / (K / 4);
        int c4 = (idx % (K / 4)) * 4;
        int gr = row0 + r;
        float4 v = make_float4(0.f, 0.f, 0.f, 0.f);
        if (gr < nrows) v = *(const float4*)(H + (size_t)gr * K + c4);
        *(float4*)(lds_h + r * LDA + c4) = v;
    }
    __syncthreads();
#endif

    const int laneM = lane & 15;
    const int koff  = (lane < 16) ? 0 : 2;   // f32 WMMA A/B layout: K 0/1 vs 2/3

    v8f acc[NTPW];
#pragma unroll
    for (int t = 0; t < NTPW; ++t) acc[t] = (v8f){0.f,0.f,0.f,0.f,0.f,0.f,0.f,0.f};

    for (int k0 = 0; k0 < K; k0 += 4) {
#pragma unroll
        for (int t = 0; t < NTPW; ++t) {
            int idx = wave + t * 8;
            int ct = idx % NCT, rt = idx / NCT;
            v2f b = *(const v2f*)(W + (size_t)(ct * 16 + laneM) * K + k0 + koff);
            v2f a = *(const v2f*)(lds_h + (rt * 16 + laneM) * LDA + k0 + koff);
            acc[t] = __builtin_amdgcn_wmma_f32_16x16x4_f32(
                false, a, false, b, (short)0, acc[t], false, false);
        }
    }

#pragma unroll
    for (int t = 0; t < NTPW; ++t) {
        int idx = wave + t * 8;
        int ct = idx % NCT, rt = idx / NCT;
        int n  = ct * 16 + laneM;
        int rbase = row0 + rt * 16 + ((lane < 16) ? 0 : 8);
#pragma unroll
        for (int v = 0; v < 8; ++v) {
            int r = rbase + v;
            if (r < nrows) {
                float val = acc[t][v];
                if (MODE >= 1) val += bias[n];
                if (MODE == 2) val = fmaxf(val, 0.f);
                Y[(size_t)r * NCOL + n] = val;
            }
        }
    }
}

// ---------------------------------------------------------------------------
// Scaled propagation: for each edge e (one wave32 per edge):
//   Acc[dst,:] += (ew[e] * Cvec[dst]) * M[src,:]
// float4 per lane, native global_atomic_add_f32 via unsafeAtomicAdd.
// ---------------------------------------------------------------------------
__global__ __launch_bounds__(256) void prop_scaled_kernel(const int* __restrict__ ei,
        const float* __restrict__ ew, const float* __restrict__ cvec,
        const float* __restrict__ M, float* __restrict__ Acc, int nedges) {
    long long g = (long long)blockIdx.x * blockDim.x + threadIdx.x;
    int e = (int)(g >> 5);
    int lane = (int)(g & 31);
    if (e >= nedges) return;
    int src = ei[e];
    int dst = ei[nedges + e];
    float coeff = ew[e] * cvec[dst];
    float4 m = *(const float4*)(M + (size_t)src * DBLK + lane * 4);
    float* a = Acc + (size_t)dst * DBLK + lane * 4;
    unsafeAtomicAdd(a + 0, coeff * m.x);
    unsafeAtomicAdd(a + 1, coeff * m.y);
    unsafeAtomicAdd(a + 2, coeff * m.z);
    unsafeAtomicAdd(a + 3, coeff * m.w);
}

// ---------------------------------------------------------------------------
// h_new = leaky_relu(gcn + h_res)   (in-place into gcn buffer is safe)
// ---------------------------------------------------------------------------
__global__ __launch_bounds__(256) void combine_kernel(const float* __restrict__ gcn,
        const float* __restrict__ hres, float* __restrict__ hout, long long total) {
    long long g = (long long)blockIdx.x * blockDim.x + threadIdx.x;
    if (g >= total) return;
    float v = gcn[g] + hres[g];
    hout[g] = (v > 0.f) ? v : LRELU * v;
}

// ---------------------------------------------------------------------------
// Row-wise log_softmax over 64 logits (one wave32 per row, 2 cols per lane)
// ---------------------------------------------------------------------------
__global__ __launch_bounds__(256) void logsoftmax_kernel(const float* __restrict__ logits,
                                                         float* __restrict__ out, int n) {
    long long g = (long long)blockIdx.x * blockDim.x + threadIdx.x;
    int row = (int)(g >> 5), lane = (int)(g & 31);
    if (row >= n) return;
    float2 v = *(const float2*)(logits + (size_t)row * 64 + lane * 2);
    float mx = fmaxf(v.x, v.y);
    for (int off = 16; off; off >>= 1) mx = fmaxf(mx, __shfl_xor(mx, off, 32));
    float s = __expf(v.x - mx) + __expf(v.y - mx);
    for (int off = 16; off; off >>= 1) s += __shfl_xor(s, off, 32);
    float lse = mx + __logf(s);
    *(float2*)(out + (size_t)row * 64 + lane * 2) = make_float2(v.x - lse, v.y - lse);
}

// ---------------------------------------------------------------------------
// emb = h / max(||h||_2, 1e-12)  (one wave32 per row, float4 per lane)
// ---------------------------------------------------------------------------
__global__ __launch_bounds__(256) void emb_norm_kernel(const float* __restrict__ h,
                                                       float* __restrict__ out, int n) {
    long long g = (long long)blockIdx.x * blockDim.x + threadIdx.x;
    int row = (int)(g >> 5), lane = (int)(g & 31);
    if (row >= n) return;
    float4 v = *(const float4*)(h + (size_t)row * DBLK + lane * 4);
    float ss = v.x * v.x + v.y * v.y + v.z * v.z + v.w * v.w;
    for (int off = 16; off; off >>= 1) ss += __shfl_xor(ss, off, 32);
    float inv = 1.0f / fmaxf(sqrtf(ss), 1e-12f);
    float4 o = make_float4(v.x * inv, v.y * inv, v.z * inv, v.w * inv);
    *(float4*)(out + (size_t)row * DBLK + lane * 4) = o;
}

// ---------------------------------------------------------------------------
extern "C" void kernel_launch(void* const* d_in, const int* in_sizes, int n_in,
                              void* d_out, int out_size, void* d_ws, size_t ws_size,
                              hipStream_t stream) {
    (void)n_in; (void)out_size; (void)ws_size;

    const float* x         = (const float*)d_in[0];
    const int*   ei_in     = (const int*)  d_in[1];
    const float* ew_in     = (const float*)d_in[2];
    const int*   ei_out    = (const int*)  d_in[3];
    const float* ew_out    = (const float*)d_in[4];
    const int*   ei_und    = (const int*)  d_in[5];
    const float* ew_und    = (const float*)d_in[6];
    const float* pe        = (const float*)d_in[7];
    const float* W_main_in = (const float*)d_in[8];
    const float* W_main_out= (const float*)d_in[9];
    const float* W_shared  = (const float*)d_in[10];
    const float* W_und     = (const float*)d_in[11];
    const float* b_main_in = (const float*)d_in[12];
    const float* b_main_out= (const float*)d_in[13];
    const float* b_shared_in  = (const float*)d_in[14];
    const float* b_shared_out = (const float*)d_in[15];
    const float* b_und     = (const float*)d_in[16];
    const float* C_in      = (const float*)d_in[17];
    const float* C_out     = (const float*)d_in[18];
    const float* C_all     = (const float*)d_in[19];
    const float* constant_ = (const float*)d_in[20];
    const float* Wd1       = (const float*)d_in[21];
    const float* bd1       = (const float*)d_in[22];
    const float* Wd2       = (const float*)d_in[23];
    const float* bd2       = (const float*)d_in[24];

    const int N    = in_sizes[0] / DBLK;
    const int Ein  = in_sizes[2];
    const int Eout = in_sizes[4];
    const int Eund = in_sizes[6];
    const long long ND = (long long)N * DBLK;
    const int L = 2, DD = DBLK * DBLK;

    float* ws = (float*)d_ws;
    float* hA     = ws;                 // N*128
    float* hB     = ws + ND;            // N*128  (gcn accumulator / next h)
    float* m      = ws + 2 * ND;        // N*128  (transformed messages / z)
    float* wf_in  = ws + 3 * ND;        // L*128*128 fused (main_in + shared)
    float* wf_out = wf_in + L * DD;     // L*128*128 fused (main_out + shared)

    dim3 blk(256);
    dim3 grid_ew((unsigned)((ND + 255) / 256));
    dim3 grid_gemm((unsigned)((N + 63) / 64));
    dim3 grid_row((unsigned)(((long long)N * 32 + 255) / 256));

    // positional encoding
    pe_add_kernel<<<grid_ew, blk, 0, stream>>>(x, pe, hA, ND);

    // fuse shared weights into main weights (both layers)
    wfuse_kernel<<<dim3((L * DD + 255) / 256), blk, 0, stream>>>(W_main_in,  W_shared, wf_in,  L * DD);
    wfuse_kernel<<<dim3((L * DD + 255) / 256), blk, 0, stream>>>(W_main_out, W_shared, wf_out, L * DD);

    float* h = hA;
    float* g = hB;
    for (int i = 0; i < L; ++i) {
        const float* cin  = C_in  + (size_t)i * N;
        const float* cout_= C_out + (size_t)i * N;
        const float* call = C_all + (size_t)i * N;

        // gcn <- constant + C-scaled biases (fully initializes accumulator)
        gcn_init_kernel<<<grid_ew, blk, 0, stream>>>(
            constant_ + (size_t)i * ND, cin, cout_, call,
            b_main_in + i * DBLK, b_shared_in + i * DBLK,
            b_main_out + i * DBLK, b_shared_out + i * DBLK,
            b_und + i * DBLK, g, ND);

        // in-direction: m = h @ (W_main_in + W_shared)^T ; scatter with C_in[dst]*w
        gemm_wmma_kernel<DBLK, DBLK, 0><<<grid_gemm, blk, 0, stream>>>(
            h, wf_in + (size_t)i * DD, nullptr, m, N);
        prop_scaled_kernel<<<dim3((unsigned)(((long long)Ein * 32 + 255) / 256)), blk, 0, stream>>>(
            ei_in, ew_in, cin, m, g, Ein);

        // out-direction
        gemm_wmma_kernel<DBLK, DBLK, 0><<<grid_gemm, blk, 0, stream>>>(
            h, wf_out + (size_t)i * DD, nullptr, m, N);
        prop_scaled_kernel<<<dim3((unsigned)(((long long)Eout * 32 + 255) / 256)), blk, 0, stream>>>(
            ei_out, ew_out, cout_, m, g, Eout);

        // undirected
        gemm_wmma_kernel<DBLK, DBLK, 0><<<grid_gemm, blk, 0, stream>>>(
            h, W_und + (size_t)i * DD, nullptr, m, N);
        prop_scaled_kernel<<<dim3((unsigned)(((long long)Eund * 32 + 255) / 256)), blk, 0, stream>>>(
            ei_und, ew_und, call, m, g, Eund);

        // h_new = leaky_relu(gcn + h)  (written in place into g), then swap
        combine_kernel<<<grid_ew, blk, 0, stream>>>(g, h, g, ND);
        float* t = h; h = g; g = t;
    }

    // decoder: z = relu(h @ Wd1^T + bd1)  [N,64]  -> reuse m
    gemm_wmma_kernel<DBLK, 64, 2><<<grid_gemm, blk, 0, stream>>>(h, Wd1, bd1, m, N);
    // logits = z @ Wd2^T + bd2  [N,64]  -> reuse dead ping-pong buffer g
    gemm_wmma_kernel<64, 64, 1><<<grid_gemm, blk, 0, stream>>>(m, Wd2, bd2, g, N);

    float* out = (float*)d_out;
    logsoftmax_kernel<<<grid_row, blk, 0, stream>>>(g, out, N);                    // logp [N,64]
    emb_norm_kernel<<<grid_row, blk, 0, stream>>>(h, out + (size_t)N * 64, N);     // emb  [N,128]
}